// MeanSquareWithManifoldItem_39659728011414
// MI455X (gfx1250) — compile-verified
//
#include <hip/hip_runtime.h>

#define N_DIM 8192
#define D_DIM 512

typedef __attribute__((ext_vector_type(2))) float v2f;
typedef __attribute__((ext_vector_type(8))) float v8f;

// ---------------------------------------------------------------------------
__global__ void init_accum_kernel(double* accum) {
    if (threadIdx.x == 0) { accum[0] = 0.0; accum[1] = 0.0; }
}

// ---------------------------------------------------------------------------
// out[row] = sum_k X[row][k]^2
// ---------------------------------------------------------------------------
__global__ __launch_bounds__(128) void row_sumsq_kernel(
    const float* __restrict__ X, float* __restrict__ out) {
    const int row = blockIdx.x;
    const float* p = X + (size_t)row * D_DIM;
    float s = 0.0f;
    for (int c = threadIdx.x; c < D_DIM; c += 128) {
        float x = p[c];
        s = fmaf(x, x, s);
    }
    for (int off = 16; off > 0; off >>= 1) s += __shfl_xor(s, off, 32);
    __shared__ float wsum[4];
    if ((threadIdx.x & 31) == 0) wsum[threadIdx.x >> 5] = s;
    __syncthreads();
    if (threadIdx.x == 0) out[row] = wsum[0] + wsum[1] + wsum[2] + wsum[3];
}

// ---------------------------------------------------------------------------
// Fused objective: R = U V^T via V_WMMA_F32_16X16X4_F32, accumulate
//   s1 = sum (R - M)^2 ;  s2 = sum W * (usq[i] + vsq[j] - 2 R)
// 256 threads = 8 waves (4x2), block tile 128x128, wave tile 32x64.
// Double-buffered LDS fed by global_load_async_to_lds_b128 (ASYNCcnt).
// ---------------------------------------------------------------------------
__global__ __launch_bounds__(256) void fused_obj_kernel(
    const float* __restrict__ U, const float* __restrict__ V,
    const float* __restrict__ Mm, const float* __restrict__ W,
    const float* __restrict__ usq, const float* __restrict__ vsq,
    double* __restrict__ accum) {
    constexpr int BM = 128, BN = 128, KC = 16;
    constexpr int LDSS = 20;                    // 80B row pitch: 16B aligned, conflict-free
    constexpr int BUF = BM * LDSS;              // floats per buffer (per matrix)
    constexpr int NCHUNK = D_DIM / KC;          // 32

    __shared__ float Us[2 * BUF];
    __shared__ float Vs[2 * BUF];

    const int tid  = threadIdx.x;
    const int lane = tid & 31;
    const int wave = tid >> 5;          // 0..7
    const int wr   = wave >> 1;         // 0..3  (row of 32)
    const int wc   = wave & 1;          // 0..1  (col of 64)
    const int i0   = blockIdx.y * BM;
    const int j0   = blockIdx.x * BN;

    const int lrow  = lane & 15;
    const int khalf = (lane >> 4) * 2;  // lanes 16..31 hold K+2,K+3 (ISA f32 A/B layout)

    const uint32_t usBase = (uint32_t)(size_t)&Us[0];
    const uint32_t vsBase = (uint32_t)(size_t)&Vs[0];

    // Issue the async global->LDS copies for one K-chunk (4 x b128 per thread).
    auto stage = [&](int buf, int kc0) {
        const int row = tid >> 2;          // 0..63
        const int c4  = (tid & 3) * 4;     // 0,4,8,12
#pragma unroll
        for (int rr = 0; rr < BM; rr += 64) {
            const float* gU = &U[(size_t)(i0 + row + rr) * D_DIM + kc0 + c4];
            const float* gV = &V[(size_t)(j0 + row + rr) * D_DIM + kc0 + c4];
            const uint32_t lo = (uint32_t)(((row + rr) * LDSS + c4 + buf * BUF) * 4);
            const uint32_t lU = usBase + lo;
            const uint32_t lV = vsBase + lo;
            asm volatile("global_load_async_to_lds_b128 %0, %1, off"
                         :: "v"(lU), "v"(gU) : "memory");
            asm volatile("global_load_async_to_lds_b128 %0, %1, off"
                         :: "v"(lV), "v"(gV) : "memory");
        }
    };

    v8f acc[2][4] = {};

    stage(0, 0);
    for (int c = 0; c < NCHUNK; ++c) {
        const int cur = c & 1;
        if (c + 1 < NCHUNK) {
            stage(cur ^ 1, (c + 1) * KC);
            // 8 outstanding: drain the 4 belonging to chunk c (in-order completion)
            asm volatile("s_wait_asynccnt 0x4" ::: "memory");
        } else {
            asm volatile("s_wait_asynccnt 0x0" ::: "memory");
        }
        __syncthreads();

        const int co = cur * BUF;
#pragma unroll
        for (int k0 = 0; k0 < KC; k0 += 4) {
            v2f a[2], b[4];
#pragma unroll
            for (int s = 0; s < 2; ++s)
                a[s] = *(const v2f*)&Us[co + (wr * 32 + s * 16 + lrow) * LDSS + k0 + khalf];
#pragma unroll
            for (int s = 0; s < 4; ++s)
                b[s] = *(const v2f*)&Vs[co + (wc * 64 + s * 16 + lrow) * LDSS + k0 + khalf];
#pragma unroll
            for (int si = 0; si < 2; ++si)
#pragma unroll
                for (int sj = 0; sj < 4; ++sj)
                    acc[si][sj] = __builtin_amdgcn_wmma_f32_16x16x4_f32(
                        false, a[si], false, b[sj], (short)0, acc[si][sj],
                        false, false);
        }
        __syncthreads();
    }

    // ----- fused epilogue: R never touches HBM -----
    float s1 = 0.0f, s2 = 0.0f;
#pragma unroll
    for (int si = 0; si < 2; ++si) {
#pragma unroll
        for (int sj = 0; sj < 4; ++sj) {
            const int j = j0 + wc * 64 + sj * 16 + lrow;
            const float vs = vsq[j];
#pragma unroll
            for (int g = 0; g < 8; ++g) {
                const int m = (lane < 16) ? g : (g + 8);   // C/D VGPR layout
                const int i = i0 + wr * 32 + si * 16 + m;
                const float rres = acc[si][sj][g];
                const size_t off = (size_t)i * N_DIM + j;
                const float mv = Mm[off];
                const float wv = W[off];
                const float d  = rres - mv;
                s1 = fmaf(d, d, s1);
                s2 = fmaf(wv, usq[i] + vs - 2.0f * rres, s2);
            }
        }
    }

    for (int off = 16; off > 0; off >>= 1) {
        s1 += __shfl_xor(s1, off, 32);
        s2 += __shfl_xor(s2, off, 32);
    }
    if (lane == 0) {
        atomicAdd(&accum[0], (double)s1);
        atomicAdd(&accum[1], (double)s2);
    }
}

// ---------------------------------------------------------------------------
__global__ void finalize_kernel(const double* __restrict__ accum,
                                const float* __restrict__ alpha,
                                float* __restrict__ out) {
    if (threadIdx.x == 0) {
        const double total = accum[0] + (double)alpha[0] * accum[1];
        out[0] = (float)(total / ((double)N_DIM * (double)N_DIM));
    }
}

extern "C" void kernel_launch(void* const* d_in, const int* in_sizes, int n_in,
                              void* d_out, int out_size, void* d_ws, size_t ws_size,
                              hipStream_t stream) {
    (void)in_sizes; (void)n_in; (void)out_size; (void)ws_size;
    const float* U     = (const float*)d_in[0];
    const float* V     = (const float*)d_in[1];
    const float* Mm    = (const float*)d_in[2];
    const float* W     = (const float*)d_in[3];
    const float* alpha = (const float*)d_in[4];
    float* out = (float*)d_out;

    double* accum = (double*)d_ws;                  // 2 doubles
    float*  usq   = (float*)((char*)d_ws + 16);     // N floats
    float*  vsq   = usq + N_DIM;                    // N floats

    init_accum_kernel<<<1, 32, 0, stream>>>(accum);
    row_sumsq_kernel<<<N_DIM, 128, 0, stream>>>(U, usq);
    row_sumsq_kernel<<<N_DIM, 128, 0, stream>>>(V, vsq);

    dim3 grid(N_DIM / 128, N_DIM / 128);            // 64 x 64 workgroups
    fused_obj_kernel<<<grid, 256, 0, stream>>>(U, V, Mm, W, usq, vsq, accum);

    finalize_kernel<<<1, 32, 0, stream>>>(accum, alpha, out);
}